// ImprovedDESimplE_14431090114916
// MI455X (gfx1250) — compile-verified
//
#include <hip/hip_runtime.h>

typedef __attribute__((ext_vector_type(2))) float v2f;
typedef __attribute__((ext_vector_type(8))) float v8f;

#define E_CNT   100000
#define CYCLE_C 30

__device__ __forceinline__ float4 ld4(const float* p) {
    return *reinterpret_cast<const float4*>(p);
}

__device__ __forceinline__ float4 sin4(float4 v) {
    float4 r;
    r.x = __sinf(v.x); r.y = __sinf(v.y); r.z = __sinf(v.z); r.w = __sinf(v.w);
    return r;
}

// emb[d] = sw[d] + sum_k amps[k][e][d] * sin(freq[k][e][d]*t_k + phi[k][e][d])
__device__ __forceinline__ float4 time_emb(const float* __restrict__ amps,
                                           const float* __restrict__ freq,
                                           const float* __restrict__ phi,
                                           size_t rowOff, int off,
                                           float ty, float tmo, float tda,
                                           float4 sw)
{
    const size_t HSTRIDE = (size_t)E_CNT * 64;
    float4 acc = sw;
    const float tk[3] = {ty, tmo, tda};
#pragma unroll
    for (int k = 0; k < 3; ++k) {
        const size_t o = (size_t)k * HSTRIDE + rowOff + (size_t)off;
        float4 av = ld4(amps + o);
        float4 fv = ld4(freq + o);
        float4 pv = ld4(phi  + o);
        float4 arg;
        arg.x = fv.x * tk[k] + pv.x;
        arg.y = fv.y * tk[k] + pv.y;
        arg.z = fv.z * tk[k] + pv.z;
        arg.w = fv.w * tk[k] + pv.w;
        float4 s = sin4(arg);
        acc.x += av.x * s.x;
        acc.y += av.y * s.y;
        acc.z += av.z * s.z;
        acc.w += av.w * s.w;
    }
    return acc;
}

// p[d] = h1*r1*t1 + h2*r2*t2   with r = rel*(1+time)
__device__ __forceinline__ float4 score4(float4 h1, float4 rf, float4 tv, float4 t1,
                                         float4 h2, float4 ri, float4 t2)
{
    float4 p;
    p.x = h1.x * (rf.x + rf.x * tv.x) * t1.x + h2.x * (ri.x + ri.x * tv.x) * t2.x;
    p.y = h1.y * (rf.y + rf.y * tv.y) * t1.y + h2.y * (ri.y + ri.y * tv.y) * t2.y;
    p.z = h1.z * (rf.z + rf.z * tv.z) * t1.z + h2.z * (ri.z + ri.z * tv.z) * t2.z;
    p.w = h1.w * (rf.w + rf.w * tv.w) * t1.w + h2.w * (ri.w + ri.w * tv.w) * t2.w;
    return p;
}

__global__ __launch_bounds__(256) void desimple_score_kernel(
    const int*   __restrict__ heads, const int* __restrict__ rels,
    const int*   __restrict__ tails,
    const float* __restrict__ years, const float* __restrict__ months,
    const float* __restrict__ days,  const int*  __restrict__ date_ids,
    const float* __restrict__ ent_h, const float* __restrict__ ent_t,
    const float* __restrict__ rel_f, const float* __restrict__ rel_i,
    const float* __restrict__ stw,   const float* __restrict__ rtc,
    const float* __restrict__ amps_h, const float* __restrict__ freq_h,
    const float* __restrict__ phi_h,
    const float* __restrict__ amps_t, const float* __restrict__ freq_t,
    const float* __restrict__ phi_t,
    float* __restrict__ out, int Btot)
{
    const int lane     = threadIdx.x & 31;
    const int wave     = blockIdx.x * (blockDim.x >> 5) + (threadIdx.x >> 5);
    const int elemBase = wave * 16;
    const int m        = lane & 15;   // element slot within wave
    const int hf       = lane >> 4;   // 0: K=0,1 side, 1: K=2,3 side

    int i = elemBase + m;
    if (i >= Btot) i = Btot - 1;      // select, no divergence (stores guarded)

    const int h   = heads[i];
    const int r   = rels[i];
    const int t   = tails[i];
    const int did = date_ids[i];
    const float ty  = years[i];
    const float tmo = months[i];
    const float tda = days[i];

    const size_t hoff = (size_t)h * 64;
    const size_t toff = (size_t)t * 64;
    const float* eh_h = ent_h + hoff;
    const float* et_t = ent_t + toff;
    const float* eh_t = ent_h + toff;
    const float* et_h = ent_t + hoff;
    const float* rfp  = rel_f + (size_t)r * 128;
    const float* rip  = rel_i + (size_t)r * 128;
    const float* rtp  = rtc   + (size_t)did * 128;
    const float* swp  = stw   + (size_t)(did / CYCLE_C) * 64;

    v8f acc = {0.f, 0.f, 0.f, 0.f, 0.f, 0.f, 0.f, 0.f};
    const v2f ones = {1.0f, 1.0f};

    // ---- structural half: dims [0,64) of h1/t1/h2/t2 ----
#pragma unroll 2
    for (int j = 0; j < 8; ++j) {
        const int off = j * 8 + hf * 4;      // lane m: 8j..8j+3 ; lane m+16: 8j+4..8j+7
        float4 a1 = ld4(eh_h + off);
        float4 b1 = ld4(et_t + off);
        float4 a2 = ld4(eh_t + off);
        float4 b2 = ld4(et_h + off);
        float4 f1 = ld4(rfp + off);
        float4 f2 = ld4(rip + off);
        float4 tv = ld4(rtp + off);
        float4 p  = score4(a1, f1, tv, b1, a2, f2, b2);

        v2f A;
        A.x = p.x; A.y = p.y;
        acc = __builtin_amdgcn_wmma_f32_16x16x4_f32(false, A, false, ones,
                                                    (short)0, acc, false, false);
        A.x = p.z; A.y = p.w;
        acc = __builtin_amdgcn_wmma_f32_16x16x4_f32(false, A, false, ones,
                                                    (short)0, acc, false, false);
    }

    // ---- temporal half: dims [64,128), time embeddings ----
#pragma unroll 1
    for (int j = 0; j < 8; ++j) {
        const int off = j * 8 + hf * 4;      // temporal dim offset in [0,64)
        float4 f1 = ld4(rfp + 64 + off);
        float4 f2 = ld4(rip + 64 + off);
        float4 tv = ld4(rtp + 64 + off);
        float4 sw = ld4(swp + off);

        float4 teA = time_emb(amps_h, freq_h, phi_h, hoff, off, ty, tmo, tda, sw); // te_h_heads
        float4 teB = time_emb(amps_t, freq_t, phi_t, toff, off, ty, tmo, tda, sw); // te_t_tails
        float4 teC = time_emb(amps_h, freq_h, phi_h, toff, off, ty, tmo, tda, sw); // te_h_tails
        float4 teD = time_emb(amps_t, freq_t, phi_t, hoff, off, ty, tmo, tda, sw); // te_t_heads

        float4 p = score4(teA, f1, tv, teB, teC, f2, teD);

        v2f A;
        A.x = p.x; A.y = p.y;
        acc = __builtin_amdgcn_wmma_f32_16x16x4_f32(false, A, false, ones,
                                                    (short)0, acc, false, false);
        A.x = p.z; A.y = p.w;
        acc = __builtin_amdgcn_wmma_f32_16x16x4_f32(false, A, false, ones,
                                                    (short)0, acc, false, false);
    }

    // D[m][n] is replicated across columns n. VGPR q, lanes 0-15 -> M=q,
    // lanes 16-31 -> M=8+q. Lane 0 stores scores 0..7, lane 16 stores 8..15.
    if ((lane & 15) == 0) {
        const int obase = elemBase + hf * 8;
#pragma unroll
        for (int q = 0; q < 8; ++q) {
            if (obase + q < Btot) out[obase + q] = acc[q] * 0.5f;
        }
    }
}

extern "C" void kernel_launch(void* const* d_in, const int* in_sizes, int n_in,
                              void* d_out, int out_size, void* d_ws, size_t ws_size,
                              hipStream_t stream)
{
    const int*   heads    = (const int*)  d_in[0];
    const int*   rels     = (const int*)  d_in[1];
    const int*   tails    = (const int*)  d_in[2];
    const float* years    = (const float*)d_in[3];
    const float* months   = (const float*)d_in[4];
    const float* days     = (const float*)d_in[5];
    const int*   date_ids = (const int*)  d_in[6];
    const float* ent_h    = (const float*)d_in[7];
    const float* ent_t    = (const float*)d_in[8];
    const float* rel_f    = (const float*)d_in[9];
    const float* rel_i    = (const float*)d_in[10];
    const float* stw      = (const float*)d_in[11];
    const float* rtc      = (const float*)d_in[12];
    const float* amps_h   = (const float*)d_in[13];
    const float* freq_h   = (const float*)d_in[14];
    const float* phi_h    = (const float*)d_in[15];
    const float* amps_t   = (const float*)d_in[16];
    const float* freq_t   = (const float*)d_in[17];
    const float* phi_t    = (const float*)d_in[18];

    const int B = in_sizes[0];
    const int waves = (B + 15) / 16;           // 16 elements per wave32
    const int wavesPerBlock = 8;               // 256 threads
    const int blocks = (waves + wavesPerBlock - 1) / wavesPerBlock;

    desimple_score_kernel<<<blocks, 256, 0, stream>>>(
        heads, rels, tails, years, months, days, date_ids,
        ent_h, ent_t, rel_f, rel_i, stw, rtc,
        amps_h, freq_h, phi_h, amps_t, freq_t, phi_t,
        (float*)d_out, B);
}